// RNNGARCH_34772055228850
// MI455X (gfx1250) — compile-verified
//
#include <hip/hip_runtime.h>
#include <math.h>

typedef __attribute__((ext_vector_type(2))) float v2f;
typedef __attribute__((ext_vector_type(8))) float v8f;

#define HID 64

// lane n <-> lane n+16 exchange (wave32), via ds_swizzle group-of-32:
// xor=0x10, or=0, and=0x1f -> offset 0x401F
__device__ __forceinline__ float swap16(float x) {
  return __int_as_float(__builtin_amdgcn_ds_swizzle(__float_as_int(x), 0x401F));
}

__device__ __forceinline__ float sgpr_f(float x) {
  return __int_as_float(__builtin_amdgcn_readfirstlane(__float_as_int(x)));
}

// One wave handles 16 batch rows.
// State H[t][s] (t=0..3 tiles, s=0..7): transposed h in WMMA C/D layout:
//   lanes 0-15  (n = lane)    : h[batch n][16*t + s]
//   lanes 16-31 (n = lane-16) : h[batch n][16*t + s + 8]
__global__ __launch_bounds__(32) void garch_scan(
    const float* __restrict__ res,  // (B,S)
    const float* __restrict__ wih,  // (H,2)
    const float* __restrict__ bih,  // (H,)
    const float* __restrict__ whh,  // (H,H)
    const float* __restrict__ bhh,  // (H,)
    const float* __restrict__ fcw,  // (1,H)
    const float* __restrict__ fcb,  // (1,)
    float* __restrict__ out,        // (B,S)
    int S) {
  const int lane = threadIdx.x & 31;
  const int half = lane >> 4;   // 0: lanes 0-15, 1: lanes 16-31
  const int m    = lane & 15;   // row-in-tile / batch-in-wave
  const int b    = blockIdx.x * 16 + m;
  const float* rowp = res + (size_t)b * S;

  // ---- fc weights pinned to SGPRs --------------------------------------
  float fcv[HID];
#pragma unroll
  for (int j = 0; j < HID; ++j) fcv[j] = sgpr_f(fcw[j]);
  const float fcb0 = sgpr_f(fcb[0]);

  // ---- A-operand fragments: W_hh (M=hidden_out, K=hidden_in), K ext. ---
  // A 16x4 f32 layout: lane l -> M = l%16 ; VGPR0: K = k0 + 2*(l/16),
  //                                          VGPR1: K = k0+1 + 2*(l/16)
  v2f A[4][17];
#pragma unroll
  for (int t = 0; t < 4; ++t) {
    const int i = 16 * t + m;  // hidden-out index of this lane
#pragma unroll
    for (int c = 0; c < 16; ++c) {
      A[t][c].x = whh[i * HID + 4 * c + 2 * half + 0];
      A[t][c].y = whh[i * HID + 4 * c + 2 * half + 1];
    }
    // K-chunk 16 carries the input projection + fused bias: [wih0, wih1, bias, 0]
    const float bias = bih[i] + bhh[i];
    A[t][16].x = half ? bias : wih[2 * i + 0];  // K=64 (wih0) / K=66 (bias)
    A[t][16].y = half ? 0.0f : wih[2 * i + 1];  // K=65 (wih1) / K=67 (zero)
  }

  // ---- sigma0 = unbiased variance of this batch's residual row ---------
  // lane pair (n, n+16) splits the row in two contiguous halves
  float sum = 0.f, sq = 0.f;
  {
    const float* p = rowp + half * (S >> 1);
    for (int i = 0; i < (S >> 1); i += 4) {
      float4 v = *(const float4*)(p + i);
      sum += v.x + v.y + v.z + v.w;
      sq = fmaf(v.x, v.x, fmaf(v.y, v.y, fmaf(v.z, v.z, fmaf(v.w, v.w, sq))));
    }
  }
  const float ts = sum + swap16(sum);     // total sum   (all lanes)
  const float tq = sq + swap16(sq);       // total sumsq (all lanes)
  const float mean = ts / (float)S;
  float sigma = (tq - ts * mean) / (float)(S - 1);  // per-batch, both halves

  if (!half) out[(size_t)b * S] = sigma;

  v8f H[4] = {};  // h0 = 0

  // ---- sequential scan --------------------------------------------------
  for (int t = 1; t < S; ++t) {
    const float eps = rowp[t - 1];
    const float e2  = eps * eps;

    v8f C0 = {}, C1 = {}, C2 = {}, C3 = {};
#pragma unroll
    for (int c = 0; c < 17; ++c) {
      // Build B-operand chunk (K=4c..4c+3, N=batch):
      //   VGPR0: lanes0-15 K=k0,   lanes16-31 K=k0+2
      //   VGPR1: lanes0-15 K=k0+1, lanes16-31 K=k0+3
      v2f Bf;
      if (c < 16) {
        const int tk = c >> 2;        // K tile of H
        const int g  = (4 * c) & 15;  // s-offset within tile
        if (g < 8) {
          // all four K values live in the low lane-half of H
          const float u0 = swap16(H[tk][g + 2]);
          const float u1 = swap16(H[tk][g + 3]);
          Bf.x = half ? u0 : H[tk][g + 0];
          Bf.y = half ? u1 : H[tk][g + 1];
        } else {
          // all four K values live in the high lane-half of H
          const int gb = g - 8;
          const float u0 = swap16(H[tk][gb + 0]);
          const float u1 = swap16(H[tk][gb + 1]);
          Bf.x = half ? H[tk][gb + 2] : u0;
          Bf.y = half ? H[tk][gb + 3] : u1;
        }
      } else {
        // K ext chunk: [eps^2, sigma, 1, 0]
        Bf.x = half ? 1.0f : e2;
        Bf.y = half ? 0.0f : sigma;
      }
      C0 = __builtin_amdgcn_wmma_f32_16x16x4_f32(false, A[0][c], false, Bf,
                                                 (short)0, C0, false, false);
      C1 = __builtin_amdgcn_wmma_f32_16x16x4_f32(false, A[1][c], false, Bf,
                                                 (short)0, C1, false, false);
      C2 = __builtin_amdgcn_wmma_f32_16x16x4_f32(false, A[2][c], false, Bf,
                                                 (short)0, C2, false, false);
      C3 = __builtin_amdgcn_wmma_f32_16x16x4_f32(false, A[3][c], false, Bf,
                                                 (short)0, C3, false, false);
    }
    H[0] = C0; H[1] = C1; H[2] = C2; H[3] = C3;

    // ---- sigma = softplus(fc . h + fcb) + 1e-6 -------------------------
    // lane-half-split dot: accA uses low-half hidden indices, accB high-half
    float accA = 0.f, accB = 0.f;
#pragma unroll
    for (int tt = 0; tt < 4; ++tt) {
#pragma unroll
      for (int s = 0; s < 8; ++s) {
        const float xv = H[tt][s];
        accA = fmaf(xv, fcv[16 * tt + s], accA);
        accB = fmaf(xv, fcv[16 * tt + s + 8], accB);
      }
    }
    const float dot = accA + swap16(accB) + fcb0;  // valid in lanes 0-15
    const float sp  = (dot > 20.0f) ? dot : log1pf(__expf(dot));
    const float sg  = sp + 1e-6f;

    const float sgs = swap16(sg);
    sigma = half ? sgs : sg;                      // broadcast to both halves

    if (!half) out[(size_t)b * S + t] = sg;
  }
}

extern "C" void kernel_launch(void* const* d_in, const int* in_sizes, int n_in,
                              void* d_out, int out_size, void* d_ws, size_t ws_size,
                              hipStream_t stream) {
  const float* res = (const float*)d_in[0];
  const float* wih = (const float*)d_in[1];
  const float* bih = (const float*)d_in[2];
  const float* whh = (const float*)d_in[3];
  const float* bhh = (const float*)d_in[4];
  const float* fcw = (const float*)d_in[5];
  const float* fcb = (const float*)d_in[6];
  float* out = (float*)d_out;

  const int S = 2048;                 // fixed by the reference
  const int B = in_sizes[0] / S;      // residuals is (B,S)

  garch_scan<<<B / 16, 32, 0, stream>>>(res, wih, bih, whh, bhh, fcw, fcb,
                                        out, S);
}